// ForwardBPTTRNN_68066641707156
// MI455X (gfx1250) — compile-verified
//
#include <hip/hip_runtime.h>
#include <hip/hip_bf16.h>

static constexpr int HDIM = 384;   // hidden
static constexpr int DDIM = 128;   // input
static constexpr int ODIM = 64;    // output
static constexpr int TSTEPS = 128; // timesteps
static constexpr int HH = HDIM * HDIM;
static constexpr int NTILE = (HDIM / 16) * (HDIM / 16); // 576 16x16 tiles
static constexpr float LNEPS = 1e-6f;

typedef __attribute__((ext_vector_type(2))) float v2f;
typedef __attribute__((ext_vector_type(8))) float v8f;
typedef __attribute__((ext_vector_type(4))) unsigned int v4u;
typedef __attribute__((ext_vector_type(8))) int v8i;
typedef __attribute__((ext_vector_type(4))) int v4i;

#if defined(__HIP_DEVICE_COMPILE__) && defined(__gfx1250__) && \
    __has_builtin(__builtin_amdgcn_tensor_load_to_lds) &&      \
    __has_builtin(__builtin_amdgcn_s_wait_tensorcnt)
#define USE_TDM 1
#else
#define USE_TDM 0
#endif

// ---------------------------------------------------------------------------
// TDM: DMA a 16-row x 384-col f32 panel (row stride 384) from global to LDS.
// D# layout per CDNA5 ISA ch.8 (group0 128b, group1 256b; groups 2/3 zero
// for a 2-D tensor). Issued by one wave; completion via TENSORcnt.
// ---------------------------------------------------------------------------
__device__ inline void tdm_load_panel16x384(unsigned lds_addr, const float* gaddr) {
#if USE_TDM
    unsigned long long ga = (unsigned long long)(const void*)gaddr;
    v4u g0;
    g0[0] = 1u;                                   // count=1 (valid descriptor)
    g0[1] = lds_addr;                             // LDS byte address
    g0[2] = (unsigned)(ga & 0xffffffffu);         // global_addr[31:0]
    g0[3] = (unsigned)((ga >> 32) & 0x01ffffffu)  // global_addr[56:32]
          | (2u << 30);                           // type = 2 ("image")
    v8i g1;
    const unsigned dim0 = HDIM, dim1 = HDIM;      // tensor dims (elements)
    const unsigned til0 = HDIM, til1 = 16;        // tile dims
    const unsigned str0 = HDIM;                   // dim0 stride (elements)
    g1[0] = (int)(2u << 16);                      // data_size=2 -> 4 bytes
    g1[1] = (int)((dim0 & 0xffffu) << 16);        // tensor_dim0[15:0]
    g1[2] = (int)(((dim0 >> 16) & 0xffffu) | ((dim1 & 0xffffu) << 16));
    g1[3] = (int)(((dim1 >> 16) & 0xffffu) | ((til0 & 0xffffu) << 16));
    g1[4] = (int)(til1 & 0xffffu);                // tile_dim1; tile_dim2=0
    g1[5] = (int)str0;                            // dim0_stride[31:0]
    g1[6] = 0;                                    // stride hi / dim1_stride lo
    g1[7] = 0;
    v4i gz = {0, 0, 0, 0};
#if __clang_major__ >= 23
    v8i gz8 = {0, 0, 0, 0, 0, 0, 0, 0};
    __builtin_amdgcn_tensor_load_to_lds(g0, g1, gz, gz, gz8, 0);
#else
    __builtin_amdgcn_tensor_load_to_lds(g0, g1, gz, gz, 0);
#endif
    __builtin_amdgcn_s_wait_tensorcnt(0);
#else
    (void)lds_addr; (void)gaddr;
#endif
}

// Stage the (contiguous) 16x384 panel into LDS: TDM path or cooperative copy.
__device__ inline void stage_panel(float* lds_panel, const float* gpanel) {
#if USE_TDM
    if ((threadIdx.x >> 5) == 0) {
        unsigned lds_addr = (unsigned)(unsigned long long)(void*)lds_panel;
        tdm_load_panel16x384(lds_addr, gpanel);
    }
#else
    for (int idx = threadIdx.x; idx < 16 * HDIM; idx += 256)
        lds_panel[idx] = gpanel[idx];
#endif
    __syncthreads();
}

// ---------------------------------------------------------------------------
// block-wide sum over 384 threads
// ---------------------------------------------------------------------------
__device__ inline float block_sum384(float v, float* red) {
    int i = threadIdx.x;
    red[i] = v;
    __syncthreads();
    if (i < 128) red[i] += red[i + 128] + red[i + 256];
    __syncthreads();
    for (int st = 64; st > 0; st >>= 1) {
        if (i < st) red[i] += red[i + st];
        __syncthreads();
    }
    float r = red[0];
    __syncthreads();
    return r;
}

// ---------------------------------------------------------------------------
// Fused GRU gates + new_h + gate-derivative vectors + LayerNorm readout +
// inst_delta (grad of masked MSE wrt new_h). One block of 384 threads.
// ---------------------------------------------------------------------------
__global__ __launch_bounds__(HDIM) void gate_kernel(
    const float* __restrict__ x, const float* __restrict__ y,
    const float* __restrict__ msk,
    const float* __restrict__ Wz, const float* __restrict__ Uz, const float* __restrict__ bz,
    const float* __restrict__ Wr, const float* __restrict__ Ur, const float* __restrict__ br,
    const float* __restrict__ Wa, const float* __restrict__ Ua, const float* __restrict__ ba,
    const float* __restrict__ lng, const float* __restrict__ lnb,
    const float* __restrict__ Wo, const float* __restrict__ bo,
    const float* __restrict__ h_old, float* __restrict__ h_new,
    float* __restrict__ inst_new,
    float* __restrict__ omz, float* __restrict__ dzc, float* __restrict__ daf,
    float* __restrict__ rv, float* __restrict__ cv,
    float* __restrict__ outp /* null on pass 1 */)
{
    __shared__ float sx[DDIM], sh[HDIM], srh[HDIM], red[HDIM], shn[HDIM], sdo[ODIM];
    const int i = threadIdx.x;
    if (i < DDIM) sx[i] = x[i];
    sh[i] = h_old[i];
    __syncthreads();

    // z and r gates
    float sz = bz[i], sr = br[i];
    for (int d = 0; d < DDIM; ++d) {
        float xv = sx[d];
        sz += xv * Wz[d * HDIM + i];
        sr += xv * Wr[d * HDIM + i];
    }
    for (int k = 0; k < HDIM; ++k) {
        float hv = sh[k];
        sz += hv * Uz[k * HDIM + i];
        sr += hv * Ur[k * HDIM + i];
    }
    float z = 1.0f / (1.0f + expf(-sz));
    float r = 1.0f / (1.0f + expf(-sr));
    srh[i] = r * sh[i];
    __syncthreads();

    // candidate a and new hidden state
    float sa = ba[i];
    for (int d = 0; d < DDIM; ++d) sa += sx[d] * Wa[d * HDIM + i];
    for (int k = 0; k < HDIM; ++k) sa += srh[k] * Ua[k * HDIM + i];
    float a = tanhf(sa);
    float nh = (1.0f - z) * sh[i] + z * a;

    h_new[i] = nh;
    omz[i] = 1.0f - z;
    dzc[i] = (a - sh[i]) * z * (1.0f - z);
    daf[i] = z * (1.0f - a * a);
    rv[i]  = r;
    cv[i]  = sh[i] * r * (1.0f - r);

    // LayerNorm on new_h
    const float inv_h = 1.0f / (float)HDIM;
    float mu  = block_sum384(nh, red) * inv_h;
    float dv  = nh - mu;
    float var = block_sum384(dv * dv, red) * inv_h;
    float invs = rsqrtf(var + LNEPS);
    float xhat = dv * invs;
    shn[i] = xhat * lng[i] + lnb[i];
    __syncthreads();

    // readout Dense + loss grad wrt output
    float mval = msk[0];
    if (i < ODIM) {
        float o = bo[i];
        for (int k = 0; k < HDIM; ++k) o += shn[k] * Wo[k * ODIM + i];
        if (outp) outp[i] = o;
        sdo[i] = mval * (o - y[i]);
    }
    __syncthreads();

    // backprop Dense + LayerNorm -> inst_delta
    float g = 0.0f;
    for (int j = 0; j < ODIM; ++j) g += Wo[i * ODIM + j] * sdo[j];
    float gg  = g * lng[i];
    float mgg  = block_sum384(gg, red) * inv_h;
    float mggx = block_sum384(gg * xhat, red) * inv_h;
    inst_new[i] = invs * (gg - mgg - xhat * mggx);
}

// ---------------------------------------------------------------------------
// Jacobian build: WMMA f32 16x16x4 GEMM for M = (Ur .* c_row) @ Ua, with the
// Ur row-panel staged in LDS via the Tensor Data Mover (shared by all 8 waves
// of the block), fused epilogue adds the diagonal / rank terms.
// ---------------------------------------------------------------------------
__global__ __launch_bounds__(256) void jac_kernel(
    const float* __restrict__ Uz, const float* __restrict__ Ur,
    const float* __restrict__ Ua,
    const float* __restrict__ omz, const float* __restrict__ dzc,
    const float* __restrict__ daf, const float* __restrict__ rv,
    const float* __restrict__ cv, float* __restrict__ J)
{
    __shared__ float apan[16 * HDIM];   // 24 KB: Ur rows [tm, tm+16)
    const int wave = threadIdx.x >> 5;
    const int lane = threadIdx.x & 31;
    const int tile = blockIdx.x * 8 + wave;
    const int tm = (tile / (HDIM / 16)) * 16;   // identical for all 8 waves
    const int tn = (tile % (HDIM / 16)) * 16;
    const int m  = lane & 15;        // A-row / B-col within tile
    const int kh = (lane >> 4) * 2;  // K sub-offset for this half-wave

    stage_panel(apan, Ur + tm * HDIM);

    v8f acc = {};
    for (int kk = 0; kk < HDIM; kk += 4) {
        const int k0 = kk + kh;
        v2f af = *(const v2f*)&apan[m * HDIM + k0];   // ds_load_b64
        v2f cw = *(const v2f*)&cv[k0];
        af.x *= cw.x;
        af.y *= cw.y;
        v2f bf;
        bf.x = Ua[(k0)     * HDIM + tn + m];
        bf.y = Ua[(k0 + 1) * HDIM + tn + m];
        acc = __builtin_amdgcn_wmma_f32_16x16x4_f32(
            false, af, false, bf, (short)0, acc, false, false);
    }

    const int col  = tn + m;
    const float dafq = daf[col];
    const float dzcq = dzc[col];
    const float omzq = omz[col];
    const int rbase = tm + 8 * (lane >> 4);
#pragma unroll
    for (int v = 0; v < 8; ++v) {
        const int row = rbase + v;
        float val = dafq * (rv[row] * Ua[row * HDIM + col] + acc[v])
                  + dzcq * Uz[row * HDIM + col];
        if (row == col) val += omzq;
        J[row * HDIM + col] = val;
    }
}

// ---------------------------------------------------------------------------
// P_new = P_old @ J, WMMA f32 16x16x4 with TDM-staged A panel in LDS.
// ---------------------------------------------------------------------------
__global__ __launch_bounds__(256) void matmul_kernel(
    const float* __restrict__ A, const float* __restrict__ B,
    float* __restrict__ Dst)
{
    __shared__ float apan[16 * HDIM];   // 24 KB: A rows [tm, tm+16)
    const int wave = threadIdx.x >> 5;
    const int lane = threadIdx.x & 31;
    const int tile = blockIdx.x * 8 + wave;
    const int tm = (tile / (HDIM / 16)) * 16;   // identical for all 8 waves
    const int tn = (tile % (HDIM / 16)) * 16;
    const int m  = lane & 15;
    const int kh = (lane >> 4) * 2;

    stage_panel(apan, A + tm * HDIM);

    v8f acc = {};
    for (int kk = 0; kk < HDIM; kk += 4) {
        const int k0 = kk + kh;
        v2f af = *(const v2f*)&apan[m * HDIM + k0];   // ds_load_b64
        v2f bf;
        bf.x = B[(k0)     * HDIM + tn + m];
        bf.y = B[(k0 + 1) * HDIM + tn + m];
        acc = __builtin_amdgcn_wmma_f32_16x16x4_f32(
            false, af, false, bf, (short)0, acc, false, false);
    }
    const int col = tn + m;
    const int rbase = tm + 8 * (lane >> 4);
#pragma unroll
    for (int v = 0; v < 8; ++v)
        Dst[(rbase + v) * HDIM + col] = acc[v];
}

// ---------------------------------------------------------------------------
// delta_new = J^{-1} (delta_old - inst_old). In-place Gaussian elimination on
// J (J is dead after matmul_kernel) with pivot row staged through LDS, then
// back substitution. One block of 384 threads.
// ---------------------------------------------------------------------------
__global__ __launch_bounds__(HDIM) void solve_kernel(
    float* __restrict__ S, const float* __restrict__ delta_old,
    const float* __restrict__ inst_old, float* __restrict__ delta_new)
{
    __shared__ float rhs[HDIM];
    __shared__ float prow[HDIM];
    __shared__ float pinv;
    __shared__ float xs[HDIM];
    const int i = threadIdx.x;
    rhs[i] = delta_old[i] - inst_old[i];
    __syncthreads();

    for (int k = 0; k < HDIM; ++k) {
        if (i >= k) prow[i] = S[k * HDIM + i];
        if (i == 0) pinv = 1.0f / S[k * HDIM + k];
        __syncthreads();
        if (i > k) {
            float f = S[i * HDIM + k] * pinv;
            for (int j = k + 1; j < HDIM; ++j)
                S[i * HDIM + j] -= f * prow[j];
            rhs[i] -= f * rhs[k];
        }
        __syncthreads();
    }
    for (int k = HDIM - 1; k >= 0; --k) {
        if (i == k) xs[k] = rhs[k] / S[k * HDIM + k];
        __syncthreads();
        if (i < k) rhs[i] -= S[i * HDIM + k] * xs[k];
        __syncthreads();
    }
    delta_new[i] = xs[i];
}

// ---------------------------------------------------------------------------
// delta0 = -pj1 @ (d1 - id1)
// ---------------------------------------------------------------------------
__global__ __launch_bounds__(HDIM) void delta0_kernel(
    const float* __restrict__ P, const float* __restrict__ d,
    const float* __restrict__ idl, float* __restrict__ delta0)
{
    __shared__ float v[HDIM];
    const int i = threadIdx.x;
    v[i] = d[i] - idl[i];
    __syncthreads();
    float s = 0.0f;
    for (int q = 0; q < HDIM; ++q) s += P[i * HDIM + q] * v[q];
    delta0[i] = -s;
}

// ---------------------------------------------------------------------------
// small utility kernels
// ---------------------------------------------------------------------------
__global__ void zero_kernel(float* p, int n) {
    int i = blockIdx.x * blockDim.x + threadIdx.x;
    if (i < n) p[i] = 0.0f;
}
__global__ void copy_kernel(const float* s, float* d, int n) {
    int i = blockIdx.x * blockDim.x + threadIdx.x;
    if (i < n) d[i] = s[i];
}
__global__ void eye_kernel(float* p) {
    int i = blockIdx.x * blockDim.x + threadIdx.x;
    if (i < HH) p[i] = ((i / HDIM) == (i % HDIM)) ? 1.0f : 0.0f;
}
__global__ __launch_bounds__(256) void norm_kernel(const float* v, int n, float* out) {
    __shared__ float red[256];
    float s = 0.0f;
    for (int idx = threadIdx.x; idx < n; idx += 256) { float t = v[idx]; s += t * t; }
    red[threadIdx.x] = s;
    __syncthreads();
    for (int st = 128; st > 0; st >>= 1) {
        if (threadIdx.x < st) red[threadIdx.x] += red[threadIdx.x + st];
        __syncthreads();
    }
    if (threadIdx.x == 0) *out = sqrtf(red[0]);
}
__global__ __launch_bounds__(256) void normdiff_kernel(const float* a, const float* b,
                                                       int n, float* out) {
    __shared__ float red[256];
    float s = 0.0f;
    for (int idx = threadIdx.x; idx < n; idx += 256) {
        float t = a[idx] - b[idx];
        s += t * t;
    }
    red[threadIdx.x] = s;
    __syncthreads();
    for (int st = 128; st > 0; st >>= 1) {
        if (threadIdx.x < st) red[threadIdx.x] += red[threadIdx.x + st];
        __syncthreads();
    }
    if (threadIdx.x == 0) *out = sqrtf(red[0]);
}

// ---------------------------------------------------------------------------
extern "C" void kernel_launch(void* const* d_in, const int* in_sizes, int n_in,
                              void* d_out, int out_size, void* d_ws, size_t ws_size,
                              hipStream_t stream)
{
    const float* inputs  = (const float*)d_in[0];
    const float* targets = (const float*)d_in[1];
    const float* masks   = (const float*)d_in[2];
    const float* Wz = (const float*)d_in[3];
    const float* Uz = (const float*)d_in[4];
    const float* bz = (const float*)d_in[5];
    const float* Wr = (const float*)d_in[6];
    const float* Ur = (const float*)d_in[7];
    const float* br = (const float*)d_in[8];
    const float* Wa = (const float*)d_in[9];
    const float* Ua = (const float*)d_in[10];
    const float* ba = (const float*)d_in[11];
    const float* lng = (const float*)d_in[12];
    const float* lnb = (const float*)d_in[13];
    const float* Wo = (const float*)d_in[14];
    const float* bo = (const float*)d_in[15];
    float* out = (float*)d_out;

    float* W  = (float*)d_ws;
    float* P0 = W;
    float* P1 = W + HH;
    float* Jm = W + 2 * HH;
    float* sv = W + 3 * HH;
    float* hA  = sv + 0 * HDIM;
    float* hB  = sv + 1 * HDIM;
    float* dA  = sv + 2 * HDIM;
    float* dB  = sv + 3 * HDIM;
    float* iA  = sv + 4 * HDIM;
    float* iB  = sv + 5 * HDIM;
    float* omz = sv + 6 * HDIM;
    float* dzc = sv + 7 * HDIM;
    float* daf = sv + 8 * HDIM;
    float* rv  = sv + 9 * HDIM;
    float* cvv = sv + 10 * HDIM;
    float* dl0 = sv + 11 * HDIM;

    const int nblk = NTILE / 8; // 72 blocks of 8 waves

    for (int pass = 0; pass < 2; ++pass) {
        if (pass == 0) {
            zero_kernel<<<(12 * HDIM + 255) / 256, 256, 0, stream>>>(sv, 12 * HDIM);
        } else {
            zero_kernel<<<(HDIM + 255) / 256, 256, 0, stream>>>(hA, HDIM);
            zero_kernel<<<(HDIM + 255) / 256, 256, 0, stream>>>(iA, HDIM);
            copy_kernel<<<(HDIM + 255) / 256, 256, 0, stream>>>(dl0, dA, HDIM);
        }
        eye_kernel<<<(HH + 255) / 256, 256, 0, stream>>>(P0);

        float *hc = hA, *hn = hB, *dc = dA, *dn = dB, *ic = iA, *inx = iB;
        float *Pc = P0, *Pn = P1;

        for (int t = 0; t < TSTEPS; ++t) {
            gate_kernel<<<1, HDIM, 0, stream>>>(
                inputs + t * DDIM, targets + t * ODIM, masks + t,
                Wz, Uz, bz, Wr, Ur, br, Wa, Ua, ba, lng, lnb, Wo, bo,
                hc, hn, inx, omz, dzc, daf, rv, cvv,
                pass ? (out + t * ODIM) : (float*)nullptr);
            jac_kernel<<<nblk, 256, 0, stream>>>(Uz, Ur, Ua, omz, dzc, daf, rv, cvv, Jm);
            matmul_kernel<<<nblk, 256, 0, stream>>>(Pc, Jm, Pn);
            solve_kernel<<<1, HDIM, 0, stream>>>(Jm, dc, ic, dn);
            float* tp;
            tp = hc; hc = hn;  hn  = tp;
            tp = dc; dc = dn;  dn  = tp;
            tp = ic; ic = inx; inx = tp;
            tp = Pc; Pc = Pn;  Pn  = tp;
        }

        if (pass == 0) {
            norm_kernel<<<1, 256, 0, stream>>>(dc, HDIM, out + TSTEPS * ODIM + 3);
            delta0_kernel<<<1, HDIM, 0, stream>>>(Pc, dc, ic, dl0);
            norm_kernel<<<1, 256, 0, stream>>>(dl0, HDIM, out + TSTEPS * ODIM + 2);
        } else {
            norm_kernel<<<1, 256, 0, stream>>>(Pc, HH, out + TSTEPS * ODIM + 0);
            normdiff_kernel<<<1, 256, 0, stream>>>(dc, ic, HDIM, out + TSTEPS * ODIM + 1);
        }
    }
}